// PolicyWrapper_50680614093182
// MI455X (gfx1250) — compile-verified
//
#include <hip/hip_runtime.h>
#include <hip/hip_bf16.h>

typedef __attribute__((ext_vector_type(2))) float v2f;
typedef __attribute__((ext_vector_type(8))) float v8f;

#define GCOUNT 512   // B*A flattened graphs
#define NATOM  64
#define NBOND  128
#define FDIM   128
#define NEG_MAX (-3.4028234663852886e38f)

// ---------------------------------------------------------------------------
// 16-row x 16-col output tile GEMM micro-kernel on the f32 WMMA pipe.
// X: LDS, row-major with leading dim ldx (row = M = lane&15).
// W: global, row-major [K x ldw], column index = col (per-lane).
// A layout (f32 16x4): lanes 0-15 -> V0:K=k, V1:K=k+1 ; lanes 16-31 -> K=k+2,k+3.
// B layout (f32 4x16): lanes 0-15 -> V0:K=k, V1:K=k+1 ; lanes 16-31 -> K=k+2,k+3.
// C/D layout: VGPR r -> M = r + 8*(lane>=16), N = lane&15.
// ---------------------------------------------------------------------------
__device__ __forceinline__ v8f gemm_tile16(const float* X, int ldx,
                                           const float* __restrict__ W, int ldw,
                                           int K, int col, int lane, float bias) {
  const int l15 = lane & 15;
  const int lhalf = lane >> 4;
  v8f acc;
#pragma unroll
  for (int r = 0; r < 8; ++r) acc[r] = bias;
  const float* xrow = X + (size_t)l15 * ldx;
#pragma unroll 4
  for (int k = 0; k < K; k += 4) {
    const int ka = k + 2 * lhalf;
    v2f a, b;
    a.x = xrow[ka];
    a.y = xrow[ka + 1];
    b.x = W[(size_t)ka * ldw + col];
    b.y = W[(size_t)(ka + 1) * ldw + col];
    acc = __builtin_amdgcn_wmma_f32_16x16x4_f32(false, a, false, b,
                                                (short)0, acc, false, false);
  }
  return acc;
}

// ---------------------------------------------------------------------------
// 1) Embedding gather: atom_state[g,a,:]=atom_emb[id], bond likewise.
// ---------------------------------------------------------------------------
__global__ void embed_kernel(const int* __restrict__ atom_ids,
                             const int* __restrict__ bond_ids,
                             const float* __restrict__ aemb,
                             const float* __restrict__ bemb,
                             float* __restrict__ atom_state,
                             float* __restrict__ bond_state) {
  size_t i = (size_t)blockIdx.x * blockDim.x + threadIdx.x;
  const size_t NAE = (size_t)GCOUNT * NATOM * FDIM;
  const size_t NBE = (size_t)GCOUNT * NBOND * FDIM;
  if (i < NAE) {
    int r = (int)(i >> 7), f = (int)(i & 127);
    atom_state[i] = aemb[(size_t)atom_ids[r] * FDIM + f];
  } else if (i < NAE + NBE) {
    size_t j = i - NAE;
    int r = (int)(j >> 7), f = (int)(j & 127);
    bond_state[j] = bemb[(size_t)bond_ids[r] * FDIM + f];
  }
}

// ---------------------------------------------------------------------------
// 2) Global attention pooling (nfp.GlobalUpdate), algebraically simplified:
//    ctx[h,:] = (sum_s softmax(q)[h,s] * elems[s,:]) @ Wv + bv.
//    One workgroup (192 threads) per graph. Pure VALU (tiny FLOPs).
// ---------------------------------------------------------------------------
__global__ __launch_bounds__(192) void global_update_kernel(
    const float* __restrict__ atom_state, const float* __restrict__ bond_state,
    const int* __restrict__ atom_ids, const int* __restrict__ bond_ids,
    const float* __restrict__ qb, const float* __restrict__ qw,
    const float* __restrict__ vb, const float* __restrict__ vw,
    float* __restrict__ gs, int residual) {
  __shared__ float WQ[FDIM * 8];
  __shared__ float Q[8 * 192];
  __shared__ float RED[16];
  __shared__ float POOL[8 * FDIM];
  const int g = blockIdx.x;
  const int tid = threadIdx.x;
  for (int i = tid; i < FDIM * 8; i += 192) WQ[i] = qw[i];
  __syncthreads();
  {  // q = elems @ Wq + bq, masked
    const int s = tid;
    const bool m = (s < NATOM) ? (atom_ids[(size_t)g * NATOM + s] != 0)
                               : (bond_ids[(size_t)g * NBOND + (s - NATOM)] != 0);
    const float* row = (s < NATOM)
        ? atom_state + ((size_t)g * NATOM + s) * FDIM
        : bond_state + ((size_t)g * NBOND + (s - NATOM)) * FDIM;
    float acc[8];
#pragma unroll
    for (int h = 0; h < 8; ++h) acc[h] = qb[h];
    for (int f = 0; f < FDIM; ++f) {
      const float x = row[f];
#pragma unroll
      for (int h = 0; h < 8; ++h) acc[h] += x * WQ[f * 8 + h];
    }
#pragma unroll
    for (int h = 0; h < 8; ++h) Q[h * 192 + s] = m ? acc[h] : NEG_MAX;
  }
  __syncthreads();
  if (tid < 8) {  // per-head max
    float mx = NEG_MAX;
    for (int s = 0; s < 192; ++s) mx = fmaxf(mx, Q[tid * 192 + s]);
    RED[tid] = mx;
  }
  __syncthreads();
  {
    const int s = tid;
#pragma unroll
    for (int h = 0; h < 8; ++h) Q[h * 192 + s] = expf(Q[h * 192 + s] - RED[h]);
  }
  __syncthreads();
  if (tid < 8) {  // per-head 1/sum
    float sm = 0.f;
    for (int s = 0; s < 192; ++s) sm += Q[tid * 192 + s];
    RED[8 + tid] = 1.0f / sm;
  }
  __syncthreads();
  if (tid < FDIM) {  // pooled[h,f] = sum_s attn[h,s]*elems[s,f]
    const int f = tid;
    float acc[8];
#pragma unroll
    for (int h = 0; h < 8; ++h) acc[h] = 0.f;
    for (int s = 0; s < 192; ++s) {
      const float* row = (s < NATOM)
          ? atom_state + ((size_t)g * NATOM + s) * FDIM
          : bond_state + ((size_t)g * NBOND + (s - NATOM)) * FDIM;
      const float x = row[f];
#pragma unroll
      for (int h = 0; h < 8; ++h) acc[h] += Q[h * 192 + s] * x;
    }
#pragma unroll
    for (int h = 0; h < 8; ++h) POOL[h * FDIM + f] = acc[h] * RED[8 + h];
  }
  __syncthreads();
  if (tid < FDIM) {  // ctx = pooled @ Wv + bv ; residual
    const int col = tid, h = col >> 4;
    float acc = vb[col];
    for (int f = 0; f < FDIM; ++f) acc += POOL[h * FDIM + f] * vw[(size_t)f * FDIM + col];
    const size_t o = (size_t)g * FDIM + col;
    gs[o] = (residual ? gs[o] : 0.0f) + acc;
  }
}

// ---------------------------------------------------------------------------
// 3) Edge update: bond += D2(relu(D1(concat(bond, src, tgt, gs))))
//    WG = 16 bonds of one graph; 8 waves; WMMA f32 16x16x4.
// ---------------------------------------------------------------------------
__global__ __launch_bounds__(256) void edge_update_kernel(
    const float* __restrict__ atom_state, float* __restrict__ bond_state,
    const float* __restrict__ gs, const int* __restrict__ conn,
    const float* __restrict__ b1, const float* __restrict__ w1,
    const float* __restrict__ b2, const float* __restrict__ w2) {
  __shared__ float X[16][516];  // 16 x 512, padded (516 % 64 == 4)
  __shared__ float H[16][260];  // 16 x 256, padded
  const int g = blockIdx.x >> 3;
  const int b0 = (blockIdx.x & 7) << 4;
  const int tid = threadIdx.x;
  for (int i = tid; i < 16 * 512; i += 256) {
    const int r = i >> 9, c = i & 511;
    const size_t bb = (size_t)g * NBOND + (b0 + r);
    float v;
    if (c < 128) v = bond_state[bb * FDIM + c];
    else if (c < 256) { const int a1 = conn[bb * 2 + 1];
      v = atom_state[((size_t)g * NATOM + a1) * FDIM + (c - 128)]; }
    else if (c < 384) { const int a0 = conn[bb * 2 + 0];
      v = atom_state[((size_t)g * NATOM + a0) * FDIM + (c - 256)]; }
    else v = gs[(size_t)g * FDIM + (c - 384)];
    X[r][c] = v;
  }
  __syncthreads();
  const int wave = tid >> 5, lane = tid & 31;
  const int l15 = lane & 15, lhalf = lane >> 4;
#pragma unroll
  for (int t = 0; t < 2; ++t) {  // dense1: 512 -> 256, relu
    const int col = (wave * 2 + t) * 16 + l15;
    v8f acc = gemm_tile16(&X[0][0], 516, w1, 256, 512, col, lane, b1[col]);
#pragma unroll
    for (int r = 0; r < 8; ++r) H[r + 8 * lhalf][col] = fmaxf(acc[r], 0.0f);
  }
  __syncthreads();
  {  // dense2: 256 -> 128, residual into bond_state
    const int col = wave * 16 + l15;
    v8f acc = gemm_tile16(&H[0][0], 260, w2, 128, 256, col, lane, b2[col]);
#pragma unroll
    for (int r = 0; r < 8; ++r) {
      const int m = r + 8 * lhalf;
      const size_t o = ((size_t)g * NBOND + b0 + m) * FDIM + col;
      bond_state[o] = X[m][col] + acc[r];
    }
  }
}

// ---------------------------------------------------------------------------
// 4a) Node messages: msgs = mask * D2(relu(D1(concat(src, bond, gs))))
// ---------------------------------------------------------------------------
__global__ __launch_bounds__(256) void node_msgs_kernel(
    const float* __restrict__ atom_state, const float* __restrict__ bond_state,
    const float* __restrict__ gs, const int* __restrict__ conn,
    const int* __restrict__ bond_ids,
    const float* __restrict__ b1, const float* __restrict__ w1,
    const float* __restrict__ b2, const float* __restrict__ w2,
    float* __restrict__ msgs) {
  __shared__ float X[16][388];  // 16 x 384, padded
  __shared__ float H[16][260];
  const int g = blockIdx.x >> 3;
  const int b0 = (blockIdx.x & 7) << 4;
  const int tid = threadIdx.x;
  for (int i = tid; i < 16 * 384; i += 256) {
    const int r = i / 384, c = i - r * 384;
    const size_t bb = (size_t)g * NBOND + (b0 + r);
    float v;
    if (c < 128) { const int a1 = conn[bb * 2 + 1];
      v = atom_state[((size_t)g * NATOM + a1) * FDIM + c]; }
    else if (c < 256) v = bond_state[bb * FDIM + (c - 128)];
    else v = gs[(size_t)g * FDIM + (c - 256)];
    X[r][c] = v;
  }
  __syncthreads();
  const int wave = tid >> 5, lane = tid & 31;
  const int l15 = lane & 15, lhalf = lane >> 4;
#pragma unroll
  for (int t = 0; t < 2; ++t) {  // dense1: 384 -> 256, relu
    const int col = (wave * 2 + t) * 16 + l15;
    v8f acc = gemm_tile16(&X[0][0], 388, w1, 256, 384, col, lane, b1[col]);
#pragma unroll
    for (int r = 0; r < 8; ++r) H[r + 8 * lhalf][col] = fmaxf(acc[r], 0.0f);
  }
  __syncthreads();
  {  // dense2: 256 -> 128, masked store
    const int col = wave * 16 + l15;
    v8f acc = gemm_tile16(&H[0][0], 260, w2, 128, 256, col, lane, b2[col]);
#pragma unroll
    for (int r = 0; r < 8; ++r) {
      const int bb = b0 + r + 8 * lhalf;
      const bool bm = bond_ids[(size_t)g * NBOND + bb] != 0;
      msgs[((size_t)g * NBOND + bb) * FDIM + col] = bm ? acc[r] : 0.0f;
    }
  }
}

// ---------------------------------------------------------------------------
// 4b) Deterministic segment-sum: agg[g,a,:] = sum_{b: idx0[g,b]==a} msgs[g,b,:]
//     (fixed bond order -> bit-stable across replays; no float atomics)
// ---------------------------------------------------------------------------
__global__ __launch_bounds__(256) void aggregate_kernel(
    const float* __restrict__ msgs, const int* __restrict__ conn,
    float* __restrict__ agg) {
  __shared__ int idx0s[NBOND];
  const int g = blockIdx.x;
  const int tid = threadIdx.x;
  if (tid < NBOND) idx0s[tid] = conn[((size_t)g * NBOND + tid) * 2 + 0];
  __syncthreads();
  for (int o = tid; o < NATOM * FDIM; o += 256) {
    const int a = o >> 7, f = o & 127;
    float acc = 0.f;
    for (int b = 0; b < NBOND; ++b)
      if (idx0s[b] == a) acc += msgs[((size_t)g * NBOND + b) * FDIM + f];
    agg[((size_t)g * NATOM + a) * FDIM + f] = acc;
  }
}

// ---------------------------------------------------------------------------
// 4c) Node MLP: atom += P2(relu(P1(agg))). Rows are independent -> tile any
//     16 consecutive global atom rows.
// ---------------------------------------------------------------------------
__global__ __launch_bounds__(256) void node_mlp_kernel(
    const float* __restrict__ agg, float* __restrict__ atom_state,
    const float* __restrict__ b1, const float* __restrict__ w1,
    const float* __restrict__ b2, const float* __restrict__ w2) {
  __shared__ float X[16][132];  // 16 x 128, padded
  __shared__ float H[16][260];
  const size_t row0 = (size_t)blockIdx.x * 16;
  const int tid = threadIdx.x;
  for (int i = tid; i < 16 * 128; i += 256) {
    const int r = i >> 7, c = i & 127;
    X[r][c] = agg[(row0 + r) * FDIM + c];
  }
  __syncthreads();
  const int wave = tid >> 5, lane = tid & 31;
  const int l15 = lane & 15, lhalf = lane >> 4;
#pragma unroll
  for (int t = 0; t < 2; ++t) {  // p1: 128 -> 256, relu
    const int col = (wave * 2 + t) * 16 + l15;
    v8f acc = gemm_tile16(&X[0][0], 132, w1, 256, 128, col, lane, b1[col]);
#pragma unroll
    for (int r = 0; r < 8; ++r) H[r + 8 * lhalf][col] = fmaxf(acc[r], 0.0f);
  }
  __syncthreads();
  {  // p2: 256 -> 128, residual
    const int col = wave * 16 + l15;
    v8f acc = gemm_tile16(&H[0][0], 260, w2, 128, 256, col, lane, b2[col]);
#pragma unroll
    for (int r = 0; r < 8; ++r) {
      const size_t o = (row0 + r + 8 * lhalf) * FDIM + col;
      atom_state[o] += acc[r];
    }
  }
}

// ---------------------------------------------------------------------------
// 5) Heads: value_preds[8] then masked_prior[8][63] into d_out (512 floats).
// ---------------------------------------------------------------------------
__global__ void heads_kernel(const float* __restrict__ gs,
                             const int* __restrict__ atom_full,  // [8,64,64]
                             const float* __restrict__ vb, const float* __restrict__ vw,
                             const float* __restrict__ pb, const float* __restrict__ pw,
                             float* __restrict__ out) {
  const int g = blockIdx.x * blockDim.x + threadIdx.x;
  if (g >= GCOUNT) return;
  const float* row = gs + (size_t)g * FDIM;
  float vd = vb[0], pd = pb[0];
  for (int f = 0; f < FDIM; ++f) { const float x = row[f]; vd += x * vw[f]; pd += x * pw[f]; }
  const int b = g >> 6, a = g & 63;
  if (a == 0) out[b] = vd;
  if (a >= 1) {
    bool m = false;
    const int* ids = atom_full + (size_t)g * 64;
    for (int k = 0; k < 64; ++k) m = m || (ids[k] != 0);
    out[8 + b * 63 + (a - 1)] = m ? pd : NEG_MAX;
  }
}

// ---------------------------------------------------------------------------
extern "C" void kernel_launch(void* const* d_in, const int* in_sizes, int n_in,
                              void* d_out, int out_size, void* d_ws, size_t ws_size,
                              hipStream_t stream) {
  (void)in_sizes; (void)n_in; (void)out_size; (void)ws_size;
  // inputs (jax tree-flatten order: dicts sorted by key, lists in order)
  const int* atom = (const int*)d_in[0];          // [8,64,64]
  const int* bond = (const int*)d_in[1];          // [8,64,128]
  const int* conn = (const int*)d_in[2];          // [8,64,128,2]
  const float* atom_emb = (const float*)d_in[3];  // [32,128]
  const float* bond_emb = (const float*)d_in[4];  // [16,128]
  // edges[i]: 5+i*4 + {d1.b, d1.w, d2.b, d2.w}
  // globals[j]: 17+j*4 + {q.b, q.w, v.b, v.w}
  // nodes[i]: 33+i*8 + {d1.b, d1.w, d2.b, d2.w, p1.b, p1.w, p2.b, p2.w}
  const float* pi_b = (const float*)d_in[57];
  const float* pi_w = (const float*)d_in[58];
  const float* val_b = (const float*)d_in[59];
  const float* val_w = (const float*)d_in[60];

  // workspace layout (floats)
  float* ws = (float*)d_ws;
  float* atom_state = ws;                                   // 512*64*128
  float* bond_state = atom_state + (size_t)GCOUNT * NATOM * FDIM;  // 512*128*128
  float* gs         = bond_state + (size_t)GCOUNT * NBOND * FDIM;  // 512*128
  float* msgs       = gs + (size_t)GCOUNT * FDIM;                  // 512*128*128
  float* agg        = msgs + (size_t)GCOUNT * NBOND * FDIM;        // 512*64*128

  const size_t embed_elems = (size_t)GCOUNT * NATOM * FDIM + (size_t)GCOUNT * NBOND * FDIM;
  embed_kernel<<<(int)((embed_elems + 255) / 256), 256, 0, stream>>>(
      atom, bond, atom_emb, bond_emb, atom_state, bond_state);

  {  // globals[0], gs = ctx
    const float* gq_b = (const float*)d_in[17];
    const float* gq_w = (const float*)d_in[18];
    const float* gv_b = (const float*)d_in[19];
    const float* gv_w = (const float*)d_in[20];
    global_update_kernel<<<GCOUNT, 192, 0, stream>>>(
        atom_state, bond_state, atom, bond, gq_b, gq_w, gv_b, gv_w, gs, 0);
  }

  for (int i = 0; i < 3; ++i) {
    const float* ed1b = (const float*)d_in[5 + i * 4 + 0];
    const float* ed1w = (const float*)d_in[5 + i * 4 + 1];
    const float* ed2b = (const float*)d_in[5 + i * 4 + 2];
    const float* ed2w = (const float*)d_in[5 + i * 4 + 3];
    edge_update_kernel<<<GCOUNT * (NBOND / 16), 256, 0, stream>>>(
        atom_state, bond_state, gs, conn, ed1b, ed1w, ed2b, ed2w);

    const float* nd1b = (const float*)d_in[33 + i * 8 + 0];
    const float* nd1w = (const float*)d_in[33 + i * 8 + 1];
    const float* nd2b = (const float*)d_in[33 + i * 8 + 2];
    const float* nd2w = (const float*)d_in[33 + i * 8 + 3];
    const float* np1b = (const float*)d_in[33 + i * 8 + 4];
    const float* np1w = (const float*)d_in[33 + i * 8 + 5];
    const float* np2b = (const float*)d_in[33 + i * 8 + 6];
    const float* np2w = (const float*)d_in[33 + i * 8 + 7];
    node_msgs_kernel<<<GCOUNT * (NBOND / 16), 256, 0, stream>>>(
        atom_state, bond_state, gs, conn, bond, nd1b, nd1w, nd2b, nd2w, msgs);
    aggregate_kernel<<<GCOUNT, 256, 0, stream>>>(msgs, conn, agg);
    node_mlp_kernel<<<GCOUNT * NATOM / 16, 256, 0, stream>>>(
        agg, atom_state, np1b, np1w, np2b, np2w);

    const float* gq_b = (const float*)d_in[17 + (i + 1) * 4 + 0];
    const float* gq_w = (const float*)d_in[17 + (i + 1) * 4 + 1];
    const float* gv_b = (const float*)d_in[17 + (i + 1) * 4 + 2];
    const float* gv_w = (const float*)d_in[17 + (i + 1) * 4 + 3];
    global_update_kernel<<<GCOUNT, 192, 0, stream>>>(
        atom_state, bond_state, atom, bond, gq_b, gq_w, gv_b, gv_w, gs, 1);
  }

  heads_kernel<<<2, 256, 0, stream>>>(gs, atom, val_b, val_w, pi_b, pi_w,
                                      (float*)d_out);
}